// HeterogeneousGraph_34522947125476
// MI455X (gfx1250) — compile-verified
//
#include <hip/hip_runtime.h>

// ---------------------------------------------------------------------------
// HeterogeneousGraph SAGE for MI455X (gfx1250, wave32, WMMA)
//
// out[j] = bias_j + x_j @ B_j^T + sum_i mean_ij @ A_ij^T
//   A_ij  = linW[j][:, i*64:(i+1)*64] @ Wl[i,j]          (64x64)
//   B_j   = sum_i linW[j][:, i*64:(i+1)*64] @ Wr[i,j]    (64x64)
//   bias_j= linb_j + sum_i linW[j][:, i*64:(i+1)*64] @ bl[i,j]
// mean_ij = segment_mean over edges e{i}{j} of x_i rows.
// ---------------------------------------------------------------------------

#define CN 3
#define NNODES 100000
#define NEDGES 1000000
#define DF 64

typedef __attribute__((ext_vector_type(2))) float v2f;
typedef __attribute__((ext_vector_type(8))) float v8f;

// --------------------------- small precompute ------------------------------
// 12 blocks: blocks 0..8 -> A_ij, blocks 9..11 -> B_j + bias_j
__global__ __launch_bounds__(256)
void hg_precompute(const float* __restrict__ Wl,    // [3][3][64][64]
                   const float* __restrict__ bl,    // [3][3][64]
                   const float* __restrict__ Wr,    // [3][3][64][64]
                   const float* __restrict__ linW,  // [3][64][192]
                   const float* __restrict__ linb,  // [3][64]
                   float* __restrict__ Amat,        // [9][64][64]
                   float* __restrict__ Bmat,        // [3][64][64]
                   float* __restrict__ bias)        // [3][64]
{
    int blk = blockIdx.x;
    if (blk < 9) {
        int i = blk / 3, j = blk % 3;
        const float* L = linW + (size_t)j * 64 * 192 + i * 64; // L[n][h], stride 192
        const float* W = Wl + (size_t)(i * 3 + j) * 64 * 64;   // W[h][k]
        float* A = Amat + (size_t)blk * 64 * 64;
        for (int idx = threadIdx.x; idx < 64 * 64; idx += blockDim.x) {
            int n = idx >> 6, k = idx & 63;
            float s = 0.0f;
            for (int h = 0; h < 64; ++h) s += L[n * 192 + h] * W[h * 64 + k];
            A[idx] = s;
        }
    } else {
        int j = blk - 9;
        float* B = Bmat + (size_t)j * 64 * 64;
        for (int idx = threadIdx.x; idx < 64 * 64; idx += blockDim.x) {
            int n = idx >> 6, k = idx & 63;
            float s = 0.0f;
            for (int i = 0; i < 3; ++i) {
                const float* L = linW + (size_t)j * 64 * 192 + i * 64;
                const float* W = Wr + (size_t)(i * 3 + j) * 64 * 64;
                for (int h = 0; h < 64; ++h) s += L[n * 192 + h] * W[h * 64 + k];
            }
            B[idx] = s;
        }
        for (int n = threadIdx.x; n < 64; n += blockDim.x) {
            float s = linb[j * 64 + n];
            for (int i = 0; i < 3; ++i) {
                const float* L = linW + (size_t)j * 64 * 192 + i * 64;
                const float* bv = bl + (size_t)(i * 3 + j) * 64;
                for (int h = 0; h < 64; ++h) s += L[n * 192 + h] * bv[h];
            }
            bias[j * 64 + n] = s;
        }
    }
}

// ------------------------------- zeroing -----------------------------------
__global__ __launch_bounds__(256)
void hg_zero(float* __restrict__ p, long n)
{
    long t = (long)blockIdx.x * blockDim.x + threadIdx.x;
    if (t < n) p[t] = 0.0f;
}

// ------------------------- edge scatter (L2 atomics) -----------------------
// one (edge, 16B-chunk) per thread: E*16 threads, float4 gather + 4 f32 atomics
__global__ __launch_bounds__(256)
void hg_scatter(const float* __restrict__ x,   // [N][64]
                const int* __restrict__ ei,    // [2][E]
                float* __restrict__ accum,     // [N][64]
                float* __restrict__ cnt)       // [N]
{
    long t = (long)blockIdx.x * blockDim.x + threadIdx.x;
    int e = (int)(t >> 4);
    int c = (int)(t & 15);
    if (e >= NEDGES) return;
    int src = ei[e];
    int dst = ei[NEDGES + e];
    const float4 v = *(const float4*)(x + (size_t)src * DF + c * 4);
    float* a = accum + (size_t)dst * DF + c * 4;
    atomicAdd(a + 0, v.x);
    atomicAdd(a + 1, v.y);
    atomicAdd(a + 2, v.z);
    atomicAdd(a + 3, v.w);
    if (c == 0) atomicAdd(cnt + dst, 1.0f);
}

// ---------------- WMMA GEMM: out = bias + X @ Bm^T (init pass) -------------
// one 16x16 tile per wave; tiles = (N/16)*4 = 25000 -> 3125 blocks of 256
__global__ __launch_bounds__(256)
void hg_gemm_init(const float* __restrict__ X,    // [N][64]
                  const float* __restrict__ Bm,   // [64][64]  Bm[n][k]
                  const float* __restrict__ bias, // [64]
                  float* __restrict__ out)        // [N][64]
{
    int w = (blockIdx.x * blockDim.x + threadIdx.x) >> 5;
    int lane = threadIdx.x & 31;
    int row0 = (w >> 2) << 4;
    int col0 = (w & 3) << 4;
    int idx15 = lane & 15;           // A: row-in-tile / B: col-in-tile / D: N
    int hi2 = (lane >> 4) << 1;      // K sub-offset (0 or 2)

    const float* xr = X + (size_t)(row0 + idx15) * DF + hi2;
    const float* br = Bm + (size_t)(col0 + idx15) * DF + hi2;

    float bv = bias[col0 + idx15];   // bias depends on N only -> same all 8 VGPRs
    v8f acc = {bv, bv, bv, bv, bv, bv, bv, bv};

#pragma unroll
    for (int k = 0; k < DF; k += 4) {
        v2f a = *(const v2f*)(xr + k);
        v2f b = *(const v2f*)(br + k);
        acc = __builtin_amdgcn_wmma_f32_16x16x4_f32(false, a, false, b,
                                                    (short)0, acc, false, false);
    }

    int mhi = (lane >> 4) << 3;      // D: M = r + (lane>=16 ? 8 : 0)
    float* o = out + (size_t)row0 * DF + col0 + idx15;
#pragma unroll
    for (int r = 0; r < 8; ++r)
        o[(size_t)(r + mhi) * DF] = acc[r];
}

// ------------- WMMA GEMM: out += (accum/max(cnt,1)) @ Am^T -----------------
__global__ __launch_bounds__(256)
void hg_gemm_mean_accum(const float* __restrict__ accum, // [N][64]
                        const float* __restrict__ cnt,   // [N]
                        const float* __restrict__ Am,    // [64][64] Am[n][k]
                        float* __restrict__ out)         // [N][64]
{
    int w = (blockIdx.x * blockDim.x + threadIdx.x) >> 5;
    int lane = threadIdx.x & 31;
    int row0 = (w >> 2) << 4;
    int col0 = (w & 3) << 4;
    int idx15 = lane & 15;
    int hi2 = (lane >> 4) << 1;

    const float* ar = accum + (size_t)(row0 + idx15) * DF + hi2;
    const float* br = Am + (size_t)(col0 + idx15) * DF + hi2;

    float cv = cnt[row0 + idx15];            // row is lane-invariant over K loop
    float inv = 1.0f / fmaxf(cv, 1.0f);

    v8f acc = {0.f, 0.f, 0.f, 0.f, 0.f, 0.f, 0.f, 0.f};

#pragma unroll
    for (int k = 0; k < DF; k += 4) {
        v2f a = *(const v2f*)(ar + k);
        a.x *= inv;
        a.y *= inv;
        v2f b = *(const v2f*)(br + k);
        acc = __builtin_amdgcn_wmma_f32_16x16x4_f32(false, a, false, b,
                                                    (short)0, acc, false, false);
    }

    int mhi = (lane >> 4) << 3;
    float* o = out + (size_t)row0 * DF + col0 + idx15;
#pragma unroll
    for (int r = 0; r < 8; ++r)
        o[(size_t)(r + mhi) * DF] += acc[r];   // wave owns tile; launches serial on stream
}

// ------------------------------- launcher ----------------------------------
extern "C" void kernel_launch(void* const* d_in, const int* in_sizes, int n_in,
                              void* d_out, int out_size, void* d_ws, size_t ws_size,
                              hipStream_t stream)
{
    (void)in_sizes; (void)n_in; (void)out_size; (void)ws_size;

    const float* x[CN] = {(const float*)d_in[0], (const float*)d_in[1], (const float*)d_in[2]};
    const int* ei[CN][CN];
    for (int i = 0; i < CN; ++i)
        for (int j = 0; j < CN; ++j)
            ei[i][j] = (const int*)d_in[3 + i * 3 + j];
    const float* Wl   = (const float*)d_in[12];
    const float* bl   = (const float*)d_in[13];
    const float* Wr   = (const float*)d_in[14];
    const float* linW = (const float*)d_in[15];
    const float* linb = (const float*)d_in[16];
    float* out = (float*)d_out;

    // workspace layout (floats)
    float* ws    = (float*)d_ws;
    float* Amat  = ws;                         // 9*4096  = 36864
    float* Bmat  = ws + 36864;                 // 3*4096  = 12288
    float* bias  = ws + 36864 + 12288;         // 192
    float* accum = ws + 65536;                 // N*64    = 6,400,000
    float* cnt   = accum + (size_t)NNODES * DF; // N (contiguous after accum)

    hg_precompute<<<12, 256, 0, stream>>>(Wl, bl, Wr, linW, linb, Amat, Bmat, bias);

    const int gemmBlocks = ((NNODES / 16) * 4) / 8;  // 25000 waves / 8 per block = 3125
    for (int j = 0; j < CN; ++j)
        hg_gemm_init<<<gemmBlocks, 256, 0, stream>>>(
            x[j], Bmat + (size_t)j * 4096, bias + j * 64, out + (size_t)j * NNODES * DF);

    const long zn = (long)NNODES * DF + NNODES;      // accum + cnt contiguous
    const int zBlocks = (int)((zn + 255) / 256);
    const int sBlocks = (NEDGES * 16) / 256;         // 62500

    for (int i = 0; i < CN; ++i) {
        for (int j = 0; j < CN; ++j) {
            hg_zero<<<zBlocks, 256, 0, stream>>>(accum, zn);
            hg_scatter<<<sBlocks, 256, 0, stream>>>(x[i], ei[i][j], accum, cnt);
            hg_gemm_mean_accum<<<gemmBlocks, 256, 0, stream>>>(
                accum, cnt, Amat + (size_t)(i * 3 + j) * 4096, out + (size_t)j * NNODES * DF);
        }
    }
}